// ChunkedChebyshev_81166291960317
// MI455X (gfx1250) — compile-verified
//
#include <hip/hip_runtime.h>
#include <cstdint>
#include <cstddef>

typedef __attribute__((ext_vector_type(2))) float v2f;
typedef __attribute__((ext_vector_type(8))) float v8f;

#define D_FEAT 128

// ---------------------------------------------------------------------------
// Bucketing kernels: group edges by 16-row destination block.
// Segments are padded to a multiple of 8 edges; padding entries are zero
// (val = 0 contributes nothing through the A matrix).
// ---------------------------------------------------------------------------
__global__ void cheb_zero_counts(int* cnt, int* gcursor, int nb) {
    int i = blockIdx.x * blockDim.x + threadIdx.x;
    if (i < nb) cnt[i] = 0;
    if (i == 0) *gcursor = 0;
}

__global__ void cheb_zerofill_edges(float* __restrict__ sVal, int* __restrict__ sPacked, int n) {
    int i = blockIdx.x * blockDim.x + threadIdx.x;
    if (i < n) {
        sVal[i] = 0.0f;
        sPacked[i] = 0;
    }
}

__global__ void cheb_hist(const int* __restrict__ rows, int* __restrict__ cnt, int nnz) {
    int e = blockIdx.x * blockDim.x + threadIdx.x;
    if (e < nnz) atomicAdd(&cnt[rows[e] >> 4], 1);
}

__global__ void cheb_alloc(const int* __restrict__ cnt, int* __restrict__ start,
                           int* __restrict__ cursor, int* gcursor, int nb) {
    int b = blockIdx.x * blockDim.x + threadIdx.x;
    if (b < nb) {
        int padded = (cnt[b] + 7) & ~7;           // multiple of 8 -> all starts 8-aligned
        int s = atomicAdd(gcursor, padded);
        start[b] = s;
        cursor[b] = s;
    }
}

__global__ void cheb_scatter(const int* __restrict__ rows, const int* __restrict__ cols,
                             const float* __restrict__ vals, int* __restrict__ cursor,
                             float* __restrict__ sVal, int* __restrict__ sPacked, int nnz) {
    int e = blockIdx.x * blockDim.x + threadIdx.x;
    if (e < nnz) {
        int r = rows[e];
        int b = r >> 4;
        int pos = atomicAdd(&cursor[b], 1);
        sVal[pos] = vals[e];
        // col fits in 16 bits (N = 50000 < 65536); local row (0..15) in bits 19:16
        sPacked[pos] = (cols[e] & 0xFFFF) | ((r & 15) << 16);
    }
}

// ---------------------------------------------------------------------------
// WMMA SpMM: one workgroup (4 waves) owns a 16-row block x 128 features.
// Wave w accumulates feature slices {16*w} and {16*(w+4)}.
// Per 4-edge group: A(16x4)[m][k] = (owner(k)==m) ? val_k : 0,
//                   B(4x16)[k][n] = X[col_k][slice*16 + n]
// so D = A*B + C accumulates val_k * X[col_k] into the owning rows exactly.
// Segments are 8-padded with zero edges -> no tail guards, EXEC all-ones.
// ---------------------------------------------------------------------------
__global__ __launch_bounds__(128) void cheb_spmm_wmma(
    const float* __restrict__ Xin, float* __restrict__ Yout,
    const float* __restrict__ sVal, const int* __restrict__ sPacked,
    const int* __restrict__ start, const int* __restrict__ cnt, int N)
{
    const int b    = blockIdx.x;
    const int wave = threadIdx.x >> 5;   // 0..3
    const int lane = threadIdx.x & 31;
    const int lo   = lane & 15;
    const int hi   = lane >> 4;          // 0 or 1
    const int kb   = hi << 1;            // K-pair base: lanes0-15 -> K{0,1}, lanes16-31 -> K{2,3}
    const int fo0  = wave * 16 + lo;           // feature offset, slice s0 = wave
    const int fo1  = (wave + 4) * 16 + lo;     // feature offset, slice s1 = wave+4

    v8f acc0 = {};
    v8f acc1 = {};

    const int segStart = start[b];
    const int pcnt     = (cnt[b] + 7) & ~7;
    const int segEnd   = segStart + pcnt;

    for (int e = segStart; e < segEnd; e += 8) {   // uniform loop: EXEC all-ones at WMMA
        // two 4-edge groups; 8-byte aligned paired loads
        const int e0 = e + kb;
        const int e1 = e0 + 4;
        const float2 va = *(const float2*)(sVal + e0);
        const float2 vb = *(const float2*)(sVal + e1);
        const int2   pa = *(const int2*)(sPacked + e0);
        const int2   pb = *(const int2*)(sPacked + e1);

        const int ca0 = (pa.x & 0xFFFF) * D_FEAT, ra0 = pa.x >> 16;
        const int ca1 = (pa.y & 0xFFFF) * D_FEAT, ra1 = pa.y >> 16;
        const int cb0 = (pb.x & 0xFFFF) * D_FEAT, rb0 = pb.x >> 16;
        const int cb1 = (pb.y & 0xFFFF) * D_FEAT, rb1 = pb.y >> 16;

        v2f A0, A1;
        A0.x = (ra0 == lo) ? va.x : 0.0f;
        A0.y = (ra1 == lo) ? va.y : 0.0f;
        A1.x = (rb0 == lo) ? vb.x : 0.0f;
        A1.y = (rb1 == lo) ? vb.y : 0.0f;

        // 32-bit indexed gathers (offsets < 6.4M floats) -> scale_offset addressing
        v2f B00, B01, B10, B11;
        B00.x = Xin[ca0 + fo0];  B00.y = Xin[ca1 + fo0];
        B01.x = Xin[ca0 + fo1];  B01.y = Xin[ca1 + fo1];
        B10.x = Xin[cb0 + fo0];  B10.y = Xin[cb1 + fo0];
        B11.x = Xin[cb0 + fo1];  B11.y = Xin[cb1 + fo1];

        acc0 = __builtin_amdgcn_wmma_f32_16x16x4_f32(false, A0, false, B00,
                                                     (short)0, acc0, false, false);
        acc1 = __builtin_amdgcn_wmma_f32_16x16x4_f32(false, A0, false, B01,
                                                     (short)0, acc1, false, false);
        acc0 = __builtin_amdgcn_wmma_f32_16x16x4_f32(false, A1, false, B10,
                                                     (short)0, acc0, false, false);
        acc1 = __builtin_amdgcn_wmma_f32_16x16x4_f32(false, A1, false, B11,
                                                     (short)0, acc1, false, false);
    }

    // C/D layout: VGPR v, lanes 0-15 -> M = v; lanes 16-31 -> M = v + 8; N = lane&15
    const int rowBase = b * 16 + hi * 8;
    if (b * 16 + 16 <= N) {
        // uniform fast path: whole block in range -> unconditional clause of stores
#pragma unroll
        for (int v = 0; v < 8; ++v) {
            const int off = (rowBase + v) * D_FEAT;
            Yout[off + fo0] = acc0[v];
            Yout[off + fo1] = acc1[v];
        }
    } else {
#pragma unroll
        for (int v = 0; v < 8; ++v) {
            const int row = rowBase + v;
            if (row < N) {
                Yout[row * D_FEAT + fo0] = acc0[v];
                Yout[row * D_FEAT + fo1] = acc1[v];
            }
        }
    }
}

// ---------------------------------------------------------------------------
// acc = c0*X + c1*T1
// ---------------------------------------------------------------------------
__global__ void cheb_accinit(const float* __restrict__ X, const float* __restrict__ T1,
                             const float* __restrict__ coeffs, float* __restrict__ acc,
                             size_t n4) {
    size_t i = (size_t)blockIdx.x * blockDim.x + threadIdx.x;
    if (i >= n4) return;
    const float c0 = coeffs[0];
    const float c1 = coeffs[1];
    float4 x = ((const float4*)X)[i];
    float4 t = ((const float4*)T1)[i];
    float4 o;
    o.x = c0 * x.x + c1 * t.x;
    o.y = c0 * x.y + c1 * t.y;
    o.z = c0 * x.z + c1 * t.z;
    o.w = c0 * x.w + c1 * t.w;
    ((float4*)acc)[i] = o;
}

// ---------------------------------------------------------------------------
// Tk = 2*S - P (in place on S);  acc += c_k * Tk
// ---------------------------------------------------------------------------
__global__ void cheb_combine(float* __restrict__ S, const float* __restrict__ P,
                             const float* __restrict__ coeffs, int k,
                             float* __restrict__ acc, size_t n4) {
    size_t i = (size_t)blockIdx.x * blockDim.x + threadIdx.x;
    if (i >= n4) return;
    const float ck = coeffs[k];
    float4 s = ((float4*)S)[i];
    float4 p = ((const float4*)P)[i];
    float4 t;
    t.x = 2.0f * s.x - p.x;
    t.y = 2.0f * s.y - p.y;
    t.z = 2.0f * s.z - p.z;
    t.w = 2.0f * s.w - p.w;
    ((float4*)S)[i] = t;
    float4 a = ((float4*)acc)[i];
    a.x += ck * t.x;
    a.y += ck * t.y;
    a.z += ck * t.z;
    a.w += ck * t.w;
    ((float4*)acc)[i] = a;
}

// ---------------------------------------------------------------------------
static inline size_t align256(size_t x) { return (x + 255) & ~(size_t)255; }

extern "C" void kernel_launch(void* const* d_in, const int* in_sizes, int n_in,
                              void* d_out, int out_size, void* d_ws, size_t ws_size,
                              hipStream_t stream) {
    const int*   rows   = (const int*)d_in[0];
    const int*   cols   = (const int*)d_in[1];
    const float* vals   = (const float*)d_in[2];
    const float* X      = (const float*)d_in[3];
    const float* coeffs = (const float*)d_in[4];

    const int NNZ = in_sizes[0];
    const int N   = in_sizes[3] / D_FEAT;
    const int M   = in_sizes[4];
    const int NB  = (N + 15) / 16;
    const int NNZpad = NNZ + 8 * NB;          // worst-case padded edge count
    const size_t matBytes = (size_t)N * D_FEAT * sizeof(float);
    const size_t n4 = (size_t)N * D_FEAT / 4;

    // Carve workspace
    char* p = (char*)d_ws;
    float* matA = (float*)p;  p += align256(matBytes);
    float* matB = (float*)p;  p += align256(matBytes);
    float* matC = (float*)p;  p += align256(matBytes);
    int* cnt     = (int*)p;   p += align256((size_t)NB * sizeof(int));
    int* startA  = (int*)p;   p += align256((size_t)NB * sizeof(int));
    int* cursor  = (int*)p;   p += align256((size_t)NB * sizeof(int));
    int* gcursor = (int*)p;   p += align256(sizeof(int));
    float* sVal  = (float*)p; p += align256((size_t)NNZpad * sizeof(float));
    int* sPacked = (int*)p;   p += align256((size_t)NNZpad * sizeof(int));
    (void)ws_size;

    float* acc = (float*)d_out;
    float* bufs[3] = {matA, matB, matC};

    const int T = 256;
    const int gNB   = (NB + T) / T;              // covers NB + gcursor slot
    const int gNNZ  = (NNZ + T - 1) / T;
    const int gNNZp = (NNZpad + T - 1) / T;
    const int gN4   = (int)((n4 + T - 1) / T);

    // 1) bucket edges by 16-row block (recomputed every call; deterministic work)
    cheb_zero_counts<<<gNB, T, 0, stream>>>(cnt, gcursor, NB);
    cheb_zerofill_edges<<<gNNZp, T, 0, stream>>>(sVal, sPacked, NNZpad);
    cheb_hist<<<gNNZ, T, 0, stream>>>(rows, cnt, NNZ);
    cheb_alloc<<<(NB + T - 1) / T, T, 0, stream>>>(cnt, startA, cursor, gcursor, NB);
    cheb_scatter<<<gNNZ, T, 0, stream>>>(rows, cols, vals, cursor, sVal, sPacked, NNZ);

    // 2) T1 = L @ X ; acc = c0*X + c1*T1
    cheb_spmm_wmma<<<NB, 128, 0, stream>>>(X, matA, sVal, sPacked, startA, cnt, N);
    cheb_accinit<<<gN4, T, 0, stream>>>(X, matA, coeffs, acc, n4);

    // 3) Chebyshev recurrence with 3-buffer rotation (no zeroing needed:
    //    spmm fully overwrites its output matrix every call)
    int pi = -1;      // T_{k-2} buffer index; -1 => X
    int ci = 0;       // T_{k-1} buffer index
    int fi = 1;       // free buffer for the next SpMM result
    int gi = 2;       // second free buffer (consumed at k==2 rotation)
    for (int k = 2; k < M; ++k) {
        float* S = bufs[fi];
        const float* Tprev = (pi < 0) ? X : bufs[pi];
        cheb_spmm_wmma<<<NB, 128, 0, stream>>>(bufs[ci], S, sVal, sPacked, startA, cnt, N);
        cheb_combine<<<gN4, T, 0, stream>>>(S, Tprev, coeffs, k, acc, n4);
        int newFree = (pi >= 0) ? pi : gi;
        pi = ci;
        ci = fi;
        fi = newFree;
    }
}